// HyperSAGEConv_48550310314282
// MI455X (gfx1250) — compile-verified
//
#include <hip/hip_runtime.h>
#include <math.h>

typedef __attribute__((ext_vector_type(2))) float v2f;
typedef __attribute__((ext_vector_type(8))) float v8f;

#define NN 50000
#define MM 10000
#define DD 128

// ---------------------------------------------------------------------------
// zero fill
// ---------------------------------------------------------------------------
__global__ void hsage_zero(float* __restrict__ p, int n) {
    int i = blockIdx.x * blockDim.x + threadIdx.x;
    if (i < n) p[i] = 0.0f;
}

// ---------------------------------------------------------------------------
// C[rows x 128] = scale(A[rows x 128]) @ W[128 x 128]
// scale(r) = 1/max(cnt[r],1) when cnt != nullptr (fused segment-mean), else 1.
// Block: 256 threads = 8 waves; block tile = 16 rows x 128 cols,
// wave w -> 16x16 output tile at cols [16w, 16w+16).
// WMMA: V_WMMA_F32_16X16X4_F32, fp32 exact, K stepped by 4.
// A-frag layout (32b A 16x4): lane&15 = M, (lane>>4)*2 = K base, 2 VGPRs = K,K+1.
// B/C/D: lane&15 = N, lanes>=16 carry K+2/K+3 (B) or M+8 (C/D).
// ---------------------------------------------------------------------------
__global__ __launch_bounds__(256) void hsage_gemm128(
    const float* __restrict__ A, const float* __restrict__ W,
    const float* __restrict__ cnt, float* __restrict__ C) {
    __shared__ float As[16 * 132];   // +4 pad: stride 128 would alias all 64 banks
    __shared__ float Ws[64 * 128];   // one K-chunk of W (32 KB)

    const int tid  = threadIdx.x;
    const int wave = tid >> 5;
    const int lane = tid & 31;
    const int row0 = blockIdx.x * 16;

    // stage A tile (16 x 128), applying 1/count for fused segment-mean
    for (int i = tid; i < 16 * 128; i += 256) {
        const int r = i >> 7, k = i & 127;
        float v = A[(row0 + r) * DD + k];
        if (cnt) v *= 1.0f / fmaxf(cnt[row0 + r], 1.0f);
        As[r * 132 + k] = v;
    }

    v8f acc = {};
    const int m     = lane & 15;
    const int n     = lane & 15;
    const int khalf = (lane >> 4) << 1;   // 0 or 2
    const int n0    = wave << 4;

    for (int k0 = 0; k0 < 128; k0 += 64) {
        __syncthreads();
        for (int i = tid; i < 64 * 128; i += 256)
            Ws[i] = W[(k0 + (i >> 7)) * DD + (i & 127)];
        __syncthreads();
        #pragma unroll
        for (int kk = 0; kk < 64; kk += 4) {
            const int lk = kk + khalf;
            v2f a, b;
            a.x = As[m * 132 + k0 + lk];
            a.y = As[m * 132 + k0 + lk + 1];
            b.x = Ws[lk * 128 + n0 + n];
            b.y = Ws[(lk + 1) * 128 + n0 + n];
            acc = __builtin_amdgcn_wmma_f32_16x16x4_f32(
                false, a, false, b, (short)0, acc, false, false);
        }
    }

    const int rbase = (lane >> 4) << 3;
    #pragma unroll
    for (int r = 0; r < 8; ++r)
        C[(row0 + rbase + r) * DD + n0 + n] = acc[r];
}

// ---------------------------------------------------------------------------
// Scatter-accumulate: for incidence e, sum[idst[e], :] += src[isrc[e], :]
// and cnt[idst[e]] += 1. One wave per incidence, float4 gather (all source
// and destination rows are L2-resident: x_proj 25.6MB / e_proj 5MB << 192MB L2).
// ---------------------------------------------------------------------------
__global__ __launch_bounds__(256) void hsage_scatter(
    const float* __restrict__ src, const int* __restrict__ isrc,
    const int* __restrict__ idst, float* __restrict__ sum,
    float* __restrict__ cnt, int E) {
    const int gid  = blockIdx.x * 256 + threadIdx.x;
    const int e    = gid >> 5;
    const int lane = gid & 31;
    if (e >= E) return;
    const int rs = isrc[e];
    const int rd = idst[e];
    const float4 g = *reinterpret_cast<const float4*>(src + rs * DD + lane * 4);
    float* d = sum + rd * DD + lane * 4;
    atomicAdd(d + 0, g.x);
    atomicAdd(d + 1, g.y);
    atomicAdd(d + 2, g.z);
    atomicAdd(d + 3, g.w);
    if (lane == 0) atomicAdd(cnt + rd, 1.0f);
}

// ---------------------------------------------------------------------------
// out = relu(concat(x_proj, n_sum/max(ncnt,1)) @ W_u + b_u)
// K = 256, processed as four 64-deep LDS chunks of W_u.
// ---------------------------------------------------------------------------
__global__ __launch_bounds__(256) void hsage_final(
    const float* __restrict__ xproj, const float* __restrict__ nsum,
    const float* __restrict__ ncnt, const float* __restrict__ Wu,
    const float* __restrict__ bu, float* __restrict__ out) {
    __shared__ float As[16 * 260];   // 16 rows x 256 K, padded stride
    __shared__ float Ws[64 * 128];

    const int tid  = threadIdx.x;
    const int wave = tid >> 5;
    const int lane = tid & 31;
    const int row0 = blockIdx.x * 16;

    for (int i = tid; i < 16 * 256; i += 256) {
        const int r = i >> 8, k = i & 255;
        const int row = row0 + r;
        float v;
        if (k < 128) v = xproj[row * DD + k];
        else         v = nsum[row * DD + (k - 128)] / fmaxf(ncnt[row], 1.0f);
        As[r * 260 + k] = v;
    }

    v8f acc = {};
    const int m     = lane & 15;
    const int n     = lane & 15;
    const int khalf = (lane >> 4) << 1;
    const int n0    = wave << 4;

    for (int k0 = 0; k0 < 256; k0 += 64) {
        __syncthreads();
        for (int i = tid; i < 64 * 128; i += 256)
            Ws[i] = Wu[(k0 + (i >> 7)) * DD + (i & 127)];
        __syncthreads();
        #pragma unroll
        for (int kk = 0; kk < 64; kk += 4) {
            const int lk = kk + khalf;
            v2f a, b;
            a.x = As[m * 260 + k0 + lk];
            a.y = As[m * 260 + k0 + lk + 1];
            b.x = Ws[lk * 128 + n0 + n];
            b.y = Ws[(lk + 1) * 128 + n0 + n];
            acc = __builtin_amdgcn_wmma_f32_16x16x4_f32(
                false, a, false, b, (short)0, acc, false, false);
        }
    }

    const int rbase = (lane >> 4) << 3;
    const float bias = bu[n0 + n];
    #pragma unroll
    for (int r = 0; r < 8; ++r) {
        const float v = acc[r] + bias;
        out[(row0 + rbase + r) * DD + n0 + n] = fmaxf(v, 0.0f);
    }
}

// ---------------------------------------------------------------------------
// launch
// ---------------------------------------------------------------------------
extern "C" void kernel_launch(void* const* d_in, const int* in_sizes, int n_in,
                              void* d_out, int out_size, void* d_ws, size_t ws_size,
                              hipStream_t stream) {
    const float* x   = (const float*)d_in[0];
    const int*   ei  = (const int*)d_in[1];
    const float* Wv  = (const float*)d_in[2];
    const float* We  = (const float*)d_in[3];
    const float* Wu  = (const float*)d_in[4];
    const float* bu  = (const float*)d_in[5];
    float* out = (float*)d_out;

    const int E   = in_sizes[1] / 2;
    const int* row = ei;        // node index per incidence
    const int* col = ei + E;    // hyperedge index per incidence (min(col)==0 for this data)

    // workspace layout: xproj | eproj | [esum | ecnt | nsum | ncnt] (accumulators contiguous)
    float* xproj = (float*)d_ws;                    // NN*DD
    float* eproj = xproj + (size_t)NN * DD;         // MM*DD
    float* esum  = eproj + (size_t)MM * DD;         // MM*DD
    float* ecnt  = esum  + (size_t)MM * DD;         // MM
    float* nsum  = ecnt  + (size_t)MM;              // NN*DD
    float* ncnt  = nsum  + (size_t)NN * DD;         // NN

    const int nzero = MM * DD + MM + NN * DD + NN;  // esum..ncnt
    hsage_zero<<<(nzero + 255) / 256, 256, 0, stream>>>(esum, nzero);

    // 1) x_proj = x @ W_v                       (3125 blocks of 16 rows)
    hsage_gemm128<<<NN / 16, 256, 0, stream>>>(x, Wv, nullptr, xproj);

    // 2) node -> hyperedge scatter-accumulate   (one wave per incidence)
    hsage_scatter<<<(E * 32 + 255) / 256, 256, 0, stream>>>(xproj, row, col,
                                                            esum, ecnt, E);

    // 3) e_proj = (e_sum / cnt) @ W_e           (fused mean, 625 blocks)
    hsage_gemm128<<<MM / 16, 256, 0, stream>>>(esum, We, ecnt, eproj);

    // 4) hyperedge -> node scatter-accumulate
    hsage_scatter<<<(E * 32 + 255) / 256, 256, 0, stream>>>(eproj, col, row,
                                                            nsum, ncnt, E);

    // 5) out = relu(concat(x_proj, n_sum/cnt) @ W_u + b_u)
    hsage_final<<<NN / 16, 256, 0, stream>>>(xproj, nsum, ncnt, Wu, bu, out);
}